// NSAFlowLayer_86998857547817
// MI455X (gfx1250) — compile-verified
//
#include <hip/hip_runtime.h>
#include <hip/hip_bf16.h>
#include <math.h>

// ---------------- problem constants (from reference) ----------------
#define PP      16384
#define KK      64
#define BATCH   32
#define MAXIT   50
#define NSIT    8
#define NCHUNK  16
#define CHUNK   (PP / NCHUNK)     // 1024 rows per workgroup
#define NBLK    (CHUNK / 64)      // 16 row-blocks of 64
#define ST      68                // LDS row stride: conflict-free + 16B aligned
#define EPSV    1e-8f

typedef __attribute__((ext_vector_type(2))) float v2f;
typedef __attribute__((ext_vector_type(8))) float v8f;

__device__ __forceinline__ v8f wmma4(v2f a, v2f b, v8f c) {
  return __builtin_amdgcn_wmma_f32_16x16x4_f32(false, a, false, b, (short)0, c,
                                               false, false);
}

__device__ __forceinline__ v8f vzero8() {
  v8f z = {0.f, 0.f, 0.f, 0.f, 0.f, 0.f, 0.f, 0.f};
  return z;
}

// ---- CDNA5 async global->LDS staging (ASYNCcnt-tracked) ----
// Stage 16 consecutive floats of one row: 4 x b128, lds/global share offsets.
__device__ __forceinline__ void stage_row4(uint32_t ldsoff, uint32_t goff,
                                           uint64_t sbase) {
  asm volatile(
      "global_load_async_to_lds_b128 %0, %1, %2 offset:0\n\t"
      "global_load_async_to_lds_b128 %0, %1, %2 offset:16\n\t"
      "global_load_async_to_lds_b128 %0, %1, %2 offset:32\n\t"
      "global_load_async_to_lds_b128 %0, %1, %2 offset:48"
      :: "v"(ldsoff), "v"(goff), "s"(sbase)
      : "memory");
}
__device__ __forceinline__ void wait_async4() {
  asm volatile("s_wait_asynccnt 0x4" ::: "memory");
}
__device__ __forceinline__ void wait_async0() {
  asm volatile("s_wait_asynccnt 0x0" ::: "memory");
}
__device__ __forceinline__ uint32_t lds_off(const float* p) {
  return (uint32_t)(uintptr_t)p;  // low 32 bits of flat shared addr = LDS offset
}

// Fused tile-pair: one shared A-fragment feeds two WMMAs (tiles tj0, tj0+1).
__device__ __forceinline__ void mm_ab2(const float* A, const float* Bm, int ti,
                                       int tj0, int m, int half, v8f& c0,
                                       v8f& c1) {
#pragma unroll
  for (int k = 0; k < 64; k += 4) {
    const int kk = k + half * 2;
    v2f a, b0, b1;
    a.x = A[(ti * 16 + m) * ST + kk];
    a.y = A[(ti * 16 + m) * ST + kk + 1];
    b0.x = Bm[kk * ST + tj0 * 16 + m];
    b0.y = Bm[(kk + 1) * ST + tj0 * 16 + m];
    b1.x = Bm[kk * ST + (tj0 + 1) * 16 + m];
    b1.y = Bm[(kk + 1) * ST + (tj0 + 1) * 16 + m];
    c0 = wmma4(a, b0, c0);
    c1 = wmma4(a, b1, c1);
  }
}

// Gram pair: acc += Yn^T Yn tiles (ti,tj0) and (ti,tj0+1), shared A-fragment.
__device__ __forceinline__ void mm_ata2(const float* Yn, int ti, int tj0, int m,
                                        int half, v8f& c0, v8f& c1) {
#pragma unroll
  for (int k = 0; k < 64; k += 4) {
    const int kk = k + half * 2;
    v2f a, b0, b1;
    a.x = Yn[kk * ST + ti * 16 + m];
    a.y = Yn[(kk + 1) * ST + ti * 16 + m];
    b0.x = Yn[kk * ST + tj0 * 16 + m];
    b0.y = Yn[(kk + 1) * ST + tj0 * 16 + m];
    b1.x = Yn[kk * ST + (tj0 + 1) * 16 + m];
    b1.y = Yn[(kk + 1) * ST + (tj0 + 1) * 16 + m];
    c0 = wmma4(a, b0, c0);
    c1 = wmma4(a, b1, c1);
  }
}

// ------------------------------------------------------------------
// Kernel 1: Yg_i = 0.9*relu(Yg_{i-1} @ S') + 0.1*X0   (first: Yg_0 = X0)
//           + partial Gram(Yg_i); async double-buffered staging.
// grid = (NCHUNK, BATCH), 256 threads (8 waves)
// ------------------------------------------------------------------
__global__ __launch_bounds__(256) void k_step_gram(
    const float* __restrict__ X0, float* __restrict__ Y,
    const float* __restrict__ Sp, float* __restrict__ gpart, int first) {
  __shared__ float Adb[2][64 * ST];  // staged input blocks (ping/pong)
  __shared__ float Ynb[64 * ST];     // new-Y block (gram input, apply path)
  __shared__ float Spl[64 * ST];     // S' = 0.5(I + M^-1/2)

  const int tid = threadIdx.x;
  const int lane = tid & 31, wave = tid >> 5;
  const int m = lane & 15, half = lane >> 4;
  const int b = blockIdx.y, ch = blockIdx.x;
  const size_t base = (size_t)b * PP * KK + (size_t)ch * CHUNK * KK;
  const float* x0 = X0 + base;
  float* y = Y + base;

  const int rr = tid >> 2;        // staging row 0..63
  const int cc = (tid & 3) * 16;  // staging col base
  const uint64_t srcbase = (uint64_t)(uintptr_t)(first ? x0 : y);

  // prologue: stage block 0; stage S' when applying
  stage_row4(lds_off(&Adb[0][rr * ST + cc]), (uint32_t)((rr * KK + cc) * 4),
             srcbase);
  if (!first) {
    stage_row4(lds_off(&Spl[rr * ST + cc]), (uint32_t)((rr * 64 + cc) * 4),
               (uint64_t)(uintptr_t)(Sp + (size_t)b * 4096));
  }

  v8f acc0 = vzero8(), acc1 = vzero8();
  const int ti = wave >> 1, tj0 = (wave & 1) * 2;  // this wave's tile pair

  for (int blk = 0; blk < NBLK; ++blk) {
    const int cur = blk & 1;
    const int rowbase = blk * 64;
    __syncthreads();  // all prior readers of prefetch target are done
    if (blk + 1 < NBLK) {
      stage_row4(lds_off(&Adb[cur ^ 1][rr * ST + cc]),
                 (uint32_t)(((rowbase + 64 + rr) * KK + cc) * 4), srcbase);
      wait_async4();  // own in-order loads: block `blk` (+ S') complete
    } else {
      wait_async0();
    }
    __syncthreads();  // publish all waves' async LDS writes
    const float* Ac = Adb[cur];

    if (first) {
      // Y_0 = X0: copy staged block to global, gram directly from A[cur]
      const float4* lsrc = (const float4*)&Ac[rr * ST + cc];
      float4* gdst = (float4*)&y[(size_t)(rowbase + rr) * KK + cc];
#pragma unroll
      for (int j = 0; j < 4; ++j) gdst[j] = lsrc[j];
      mm_ata2(Ac, ti, tj0, m, half, acc0, acc1);
    } else {
      // apply: U = A[cur] @ S'; Ynew = 0.9*relu(U) + 0.1*X0
      v8f u0 = vzero8(), u1 = vzero8();
      mm_ab2(Ac, Spl, ti, tj0, m, half, u0, u1);
#pragma unroll
      for (int tt = 0; tt < 2; ++tt) {
        const v8f u = tt ? u1 : u0;
        const int ct = tj0 + tt;
#pragma unroll
        for (int r = 0; r < 8; ++r) {
          const int row = ti * 16 + r + half * 8;
          const int col = ct * 16 + m;
          const float uu = fmaxf(u[r], 0.0f);
          const float v =
              0.9f * uu + 0.1f * x0[(size_t)(rowbase + row) * KK + col];
          Ynb[row * ST + col] = v;
          y[(size_t)(rowbase + row) * KK + col] = v;
        }
      }
      __syncthreads();  // Ynb visible
      mm_ata2(Ynb, ti, tj0, m, half, acc0, acc1);
    }
  }

  // write partial gram for this chunk (tiles (ti,tj0), (ti,tj0+1))
  float* gp = gpart + ((size_t)b * NCHUNK + ch) * 4096;
#pragma unroll
  for (int r = 0; r < 8; ++r) {
    const int row = ti * 16 + r + half * 8;
    gp[row * 64 + tj0 * 16 + m] = acc0[r];
    gp[row * 64 + (tj0 + 1) * 16 + m] = acc1[r];
  }
}

// ------------------------------------------------------------------
// Kernel 2: reduce partial grams -> M; S = NS-invsqrt(M); S' = 0.5(I+S)
// grid = BATCH, 256 threads
// ------------------------------------------------------------------
__global__ __launch_bounds__(256) void k_ns(const float* __restrict__ gpart,
                                            float* __restrict__ Sp) {
  __shared__ float bufY[64 * ST];
  __shared__ float bufZ[64 * ST];
  __shared__ float bufT[64 * ST];
  __shared__ float snrm;

  const int tid = threadIdx.x;
  const int lane = tid & 31, wave = tid >> 5;
  const int m = lane & 15, half = lane >> 4;
  const int b = blockIdx.x;
  const float* gp = gpart + (size_t)b * NCHUNK * 4096;

  // deterministic partial-gram reduction
#pragma unroll
  for (int i = 0; i < 16; ++i) {
    int e = tid * 16 + i;
    float s = 0.f;
#pragma unroll
    for (int c = 0; c < NCHUNK; ++c) s += gp[(size_t)c * 4096 + e];
    bufY[(e >> 6) * ST + (e & 63)] = s;
  }
  __syncthreads();
  if (tid == 0) {
    float tr = EPSV;
    for (int i = 0; i < 64; ++i) tr += bufY[i * ST + i];
    snrm = tr;
  }
  __syncthreads();
  const float inv = 1.0f / snrm;
#pragma unroll
  for (int i = 0; i < 16; ++i) {
    int e = tid * 16 + i;
    int r = e >> 6, c = e & 63;
    bufY[r * ST + c] *= inv;
    bufZ[r * ST + c] = (r == c) ? 1.0f : 0.0f;
  }
  __syncthreads();

  const int ti = wave >> 1, tj0 = (wave & 1) * 2;
  for (int it = 0; it < NSIT; ++it) {
    // T = 1.5*I - 0.5*(Z@Y)
    v8f t0a = vzero8(), t1a = vzero8();
    mm_ab2(bufZ, bufY, ti, tj0, m, half, t0a, t1a);
#pragma unroll
    for (int tt = 0; tt < 2; ++tt) {
      const v8f c8 = tt ? t1a : t0a;
      const int tj = tj0 + tt;
#pragma unroll
      for (int r = 0; r < 8; ++r) {
        const int row = ti * 16 + r + half * 8;
        const int col = tj * 16 + m;
        bufT[row * ST + col] = ((row == col) ? 1.5f : 0.0f) - 0.5f * c8[r];
      }
    }
    __syncthreads();
    // register-staged: Ynew = Y@T, Znew = T@Z (this wave: tile pair of both)
    v8f ry0 = vzero8(), ry1 = vzero8(), rz0 = vzero8(), rz1 = vzero8();
    mm_ab2(bufY, bufT, ti, tj0, m, half, ry0, ry1);
    mm_ab2(bufT, bufZ, ti, tj0, m, half, rz0, rz1);
    __syncthreads();  // all reads of Y/Z complete
#pragma unroll
    for (int r = 0; r < 8; ++r) {
      const int row = ti * 16 + r + half * 8;
      const int c0 = tj0 * 16 + m, c1 = (tj0 + 1) * 16 + m;
      bufY[row * ST + c0] = ry0[r];
      bufY[row * ST + c1] = ry1[r];
      bufZ[row * ST + c0] = rz0[r];
      bufZ[row * ST + c1] = rz1[r];
    }
    __syncthreads();
  }

  // S' = 0.5*(I + Z / sqrt(nrm))
  const float rs = 1.0f / sqrtf(snrm);
  float* sp = Sp + (size_t)b * 4096;
#pragma unroll
  for (int i = 0; i < 16; ++i) {
    int e = tid * 16 + i;
    int r = e >> 6, c = e & 63;
    sp[e] = 0.5f * (((r == c) ? 1.0f : 0.0f) + bufZ[r * ST + c] * rs);
  }
}

// ------------------------------------------------------------------
// Kernel 3: final Y = relu(Y @ S')   (in place over d_out)
// grid = (NCHUNK, BATCH), 256 threads
// ------------------------------------------------------------------
__global__ __launch_bounds__(256) void k_final(float* __restrict__ Y,
                                               const float* __restrict__ Sp) {
  __shared__ float Spl[64 * ST];
  const int tid = threadIdx.x;
  const int lane = tid & 31, wave = tid >> 5;
  const int m = lane & 15, half = lane >> 4;
  const int b = blockIdx.y, ch = blockIdx.x;
  float* y = Y + (size_t)b * PP * KK + (size_t)ch * CHUNK * KK;

#pragma unroll
  for (int i = 0; i < 16; ++i) {
    int e = tid * 16 + i;
    Spl[(e >> 6) * ST + (e & 63)] = Sp[(size_t)b * 4096 + e];
  }
  __syncthreads();

  // 1024 rows -> 64 groups of 16 rows; 8 waves x 8 groups
  for (int g = wave; g < CHUNK / 16; g += 8) {
    const int row0 = g * 16;
    v8f c8[4];
#pragma unroll
    for (int ct = 0; ct < 4; ++ct) c8[ct] = vzero8();
#pragma unroll
    for (int k = 0; k < 64; k += 4) {
      const int kk = k + half * 2;
      v2f a;
      a.x = y[(size_t)(row0 + m) * KK + kk];
      a.y = y[(size_t)(row0 + m) * KK + kk + 1];
#pragma unroll
      for (int ct = 0; ct < 4; ++ct) {
        v2f bfr;
        bfr.x = Spl[kk * ST + ct * 16 + m];
        bfr.y = Spl[(kk + 1) * ST + ct * 16 + m];
        c8[ct] = wmma4(a, bfr, c8[ct]);
      }
    }
#pragma unroll
    for (int ct = 0; ct < 4; ++ct) {
#pragma unroll
      for (int r = 0; r < 8; ++r) {
        const int row = row0 + r + half * 8;
        const int col = ct * 16 + m;
        y[(size_t)row * KK + col] = fmaxf(c8[ct][r], 0.0f);
      }
    }
  }
}

// ------------------------------------------------------------------
extern "C" void kernel_launch(void* const* d_in, const int* in_sizes, int n_in,
                              void* d_out, int out_size, void* d_ws,
                              size_t ws_size, hipStream_t stream) {
  (void)in_sizes; (void)n_in; (void)out_size; (void)ws_size;
  const float* X0 = (const float*)d_in[0];
  float* Y = (float*)d_out;                           // Y lives in d_out
  float* gpart = (float*)d_ws;                        // B*NCHUNK*4096 f32
  float* Sp = gpart + (size_t)BATCH * NCHUNK * 4096;  // B*4096 f32

  dim3 grid1(NCHUNK, BATCH);
  for (int it = 0; it < MAXIT; ++it) {
    k_step_gram<<<grid1, 256, 0, stream>>>(X0, Y, Sp, gpart, it == 0 ? 1 : 0);
    k_ns<<<BATCH, 256, 0, stream>>>(gpart, Sp);
  }
  k_final<<<grid1, 256, 0, stream>>>(Y, Sp);
}